// SampledGNN_13683765805584
// MI455X (gfx1250) — compile-verified
//
#include <hip/hip_runtime.h>
#include <math.h>

#define N_NODES 50000
#define E_EDGES 800000
#define G_GRAPHS 64
#define HID 128
#define HEADS 4
#define HC 32
#define NLAYERS 3
#define ETOT (2 * E_EDGES + N_NODES)
#define GN_EPS 1e-5f
#define LRELU_SLOPE 0.2f

typedef float v2f __attribute__((ext_vector_type(2)));
typedef float v8f __attribute__((ext_vector_type(8)));
typedef unsigned int u32x4 __attribute__((ext_vector_type(4)));
typedef int i32x8 __attribute__((ext_vector_type(8)));

// ---------------------------------------------------------------------------
// Tensor Data Mover: flat 1-D copy of `nelem` f32 from global to LDS.
// Builds the D# (groups 0 and 1) per CDNA5 ISA ch.8 and issues
// tensor_load_to_lds via inline asm (portable across builtin-arity variants).
// ---------------------------------------------------------------------------
__device__ __forceinline__ void tdm_load_1d_f32(unsigned lds_off,
                                                const float* gptr,
                                                unsigned nelem /* <= 65535 */) {
    unsigned long long ga = (unsigned long long)(uintptr_t)gptr;
    u32x4 g0;
    g0[0] = 1u;                                            // count=1, user mode
    g0[1] = lds_off;                                       // lds_addr
    g0[2] = (unsigned)(ga & 0xFFFFFFFFu);                  // global_addr[31:0]
    g0[3] = (unsigned)((ga >> 32) & 0x01FFFFFFu)           // global_addr[56:32]
          | (2u << 30);                                    // type=2 (image)
    i32x8 g1;
    g1[0] = (int)(2u << 16);                               // data_size=2 (4B)
    g1[1] = (int)((nelem & 0xFFFFu) << 16);                // tensor_dim0[15:0]
    g1[2] = (int)(((nelem >> 16) & 0xFFFFu) | (1u << 16)); // td0[31:16] | tensor_dim1=1
    g1[3] = (int)((nelem & 0xFFFFu) << 16);                // td1[31:16]=0 | tile_dim0=nelem
    g1[4] = 1;                                             // tile_dim1=1, tile_dim2=0
    g1[5] = (int)nelem;                                    // tensor_dim0_stride[31:0]
    g1[6] = 0;                                             // stride hi / dim1_stride lo
    g1[7] = 0;
    asm volatile("tensor_load_to_lds %0, %1" :: "s"(g0), "s"(g1) : "memory");
}

// ---------------------------------------------------------------------------
// Dense GEMM: C[M,NCOL] = act(A[M,128] @ B[128,NCOL] + bias)
// One block = one 16-row M tile. TDM stages the A tile (16x128) and the whole
// weight matrix B (128xNCOL) into LDS; each wave32 then computes one 16x16
// output tile with V_WMMA_F32_16X16X4_F32 fed by constant-offset ds_loads.
// NCOL is compile-time so all LDS addressing folds to immediate offsets.
// act: 0 = none, 1 = relu.  Requires M%16==0.
// ---------------------------------------------------------------------------
template <int NCOL>
__global__ __launch_bounds__(256) void gemm_wmma_f32_t(
    const float* __restrict__ A, const float* __restrict__ B,
    const float* __restrict__ bias, float* __restrict__ C, int act)
{
    constexpr int K = HID;                // 128
    __shared__ float As[16 * K];          // 8 KB
    __shared__ float Bs[K * NCOL];        // 64 KB (NCOL=128) / 32 KB (NCOL=64)

    const int wave = threadIdx.x >> 5;
    const int lane = threadIdx.x & 31;
    const int tm   = blockIdx.x;          // M tile
    constexpr int tilesN = NCOL >> 4;

    if (wave == 0) {
        tdm_load_1d_f32((unsigned)(uintptr_t)&As[0], A + (size_t)tm * 16 * K,
                        (unsigned)(16 * K));
        tdm_load_1d_f32((unsigned)(uintptr_t)&Bs[0], B, (unsigned)(K * NCOL));
        __builtin_amdgcn_s_wait_tensorcnt(0);
    }
    __syncthreads();

    if (wave < tilesN) {                  // wave-uniform; EXEC stays full
        const int tn    = wave;
        const int l15   = lane & 15;
        const int khalf = (lane >> 4) << 1;               // 0 or 2
        const int ncol  = (tn << 4) + l15;                // B/C column
        const float* __restrict__ arow = &As[l15 * K + khalf];
        const float* __restrict__ bcol = &Bs[khalf * NCOL + ncol];

        v8f acc = {};
#pragma unroll 8
        for (int k = 0; k < K; k += 4) {
            v2f a = *(const v2f*)(arow + k);              // ds_load_b64, imm offset
            v2f b;
            b[0] = bcol[k * NCOL];                        // ds_load, imm offsets
            b[1] = bcol[k * NCOL + NCOL];
            acc = __builtin_amdgcn_wmma_f32_16x16x4_f32(
                false, a, false, b, (short)0, acc, false, false);
        }

        const float bv = bias ? bias[ncol] : 0.0f;
        const int mbase = (tm << 4) + ((lane >> 4) << 3); // lanes 0-15: M=v, 16-31: M=8+v
#pragma unroll
        for (int v = 0; v < 8; ++v) {
            float r = acc[v] + bv;
            if (act == 1) r = fmaxf(r, 0.0f);
            C[(size_t)(mbase + v) * NCOL + ncol] = r;
        }
    }
}

// ---------------------------------------------------------------------------
// Order-preserving float <-> uint map (for atomic float max)
// ---------------------------------------------------------------------------
__device__ __forceinline__ unsigned int fflip(float f) {
    unsigned int u = __float_as_uint(f);
    return (u & 0x80000000u) ? ~u : (u | 0x80000000u);
}
__device__ __forceinline__ float funflip(unsigned int u) {
    return (u & 0x80000000u) ? __uint_as_float(u & 0x7FFFFFFFu)
                             : __uint_as_float(~u);
}

__device__ __forceinline__ void edge_decode(int e, const int* __restrict__ ei0,
                                            const int* __restrict__ ei1,
                                            int& s, int& d, float ew, float& w) {
    if (e < E_EDGES)          { s = ei0[e];            d = ei1[e];            w = ew;   }
    else if (e < 2 * E_EDGES) { s = ei1[e - E_EDGES];  d = ei0[e - E_EDGES];  w = ew;   }
    else                      { s = e - 2 * E_EDGES;   d = s;                 w = 2.0f; }
}

// ---------------------------------------------------------------------------
// Fills
// ---------------------------------------------------------------------------
__global__ void fill_u32(unsigned int* p, unsigned int v, int n) {
    int t = blockIdx.x * blockDim.x + threadIdx.x;
    if (t < n) p[t] = v;
}
__global__ void fill_f32(float* p, float v, int n) {
    int t = blockIdx.x * blockDim.x + threadIdx.x;
    if (t < n) p[t] = v;
}

// g init: g[n,c] = gat_b[c] + gcn_b[c]  (both conv biases folded in)
__global__ void init_g(const float* __restrict__ gat_b,
                       const float* __restrict__ gcn_b, float* __restrict__ g) {
    int t = blockIdx.x * blockDim.x + threadIdx.x;
    if (t >= N_NODES * HID) return;
    int c = t & (HID - 1);
    g[t] = gat_b[c] + gcn_b[c];
}

// ---------------------------------------------------------------------------
// GCN degree / dinv (once per call)
// ---------------------------------------------------------------------------
__global__ void edge_deg(const int* __restrict__ ei0, const int* __restrict__ ei1,
                         const float* __restrict__ ewp, float* __restrict__ deg) {
    int e = blockIdx.x * blockDim.x + threadIdx.x;
    if (e >= ETOT) return;
    int s, d; float w;
    edge_decode(e, ei0, ei1, s, d, ewp[0], w);
    atomicAdd(&deg[d], w);
}
__global__ void node_dinv(const float* __restrict__ deg, float* __restrict__ dinv) {
    int n = blockIdx.x * blockDim.x + threadIdx.x;
    if (n >= N_NODES) return;
    float dg = deg[n];
    dinv[n] = dg > 0.0f ? rsqrtf(dg) : 0.0f;
}
__global__ void node_cnt(const int* __restrict__ batch, float* __restrict__ gcnt) {
    int n = blockIdx.x * blockDim.x + threadIdx.x;
    if (n >= N_NODES) return;
    atomicAdd(&gcnt[batch[n]], 1.0f);
}

// ---------------------------------------------------------------------------
// GAT attention scores: sc[n,h] = sum_c hlin[n, h*HC+c] * att[h,c]
// ---------------------------------------------------------------------------
__global__ void gat_scores(const float* __restrict__ hlin,
                           const float* __restrict__ asrc,
                           const float* __restrict__ adst,
                           float* __restrict__ sc_src, float* __restrict__ sc_dst) {
    int t = blockIdx.x * blockDim.x + threadIdx.x;
    if (t >= N_NODES * HEADS) return;
    int n = t >> 2, h = t & 3;
    const float* row = hlin + (size_t)n * HID + h * HC;
    float ss = 0.0f, sd = 0.0f;
#pragma unroll
    for (int c = 0; c < HC; ++c) {
        float v = row[c];
        ss += v * asrc[h * HC + c];
        sd += v * adst[h * HC + c];
    }
    sc_src[t] = ss;
    sc_dst[t] = sd;
}

__device__ __forceinline__ float leaky(float a) {
    return a > 0.0f ? a : a * LRELU_SLOPE;
}

// segment_max(alpha, dst) via flipped-uint atomicMax
__global__ void edge_amax(const int* __restrict__ ei0, const int* __restrict__ ei1,
                          const float* __restrict__ sc_src, const float* __restrict__ sc_dst,
                          unsigned int* __restrict__ amax_u) {
    int t = blockIdx.x * blockDim.x + threadIdx.x;
    if (t >= ETOT * HEADS) return;
    int e = t >> 2, h = t & 3;
    int s, d; float w;
    edge_decode(e, ei0, ei1, s, d, 1.0f, w);
    float a = leaky(sc_src[s * HEADS + h] + sc_dst[d * HEADS + h]);
    atomicMax(&amax_u[d * HEADS + h], fflip(a));
}

// denom[d,h] = segment_sum(exp(alpha - amax[d,h]))
__global__ void edge_denom(const int* __restrict__ ei0, const int* __restrict__ ei1,
                           const float* __restrict__ sc_src, const float* __restrict__ sc_dst,
                           const unsigned int* __restrict__ amax_u,
                           float* __restrict__ denom) {
    int t = blockIdx.x * blockDim.x + threadIdx.x;
    if (t >= ETOT * HEADS) return;
    int e = t >> 2, h = t & 3;
    int s, d; float w;
    edge_decode(e, ei0, ei1, s, d, 1.0f, w);
    float a = leaky(sc_src[s * HEADS + h] + sc_dst[d * HEADS + h]);
    float ex = expf(a - funflip(amax_u[d * HEADS + h]));
    atomicAdd(&denom[d * HEADS + h], ex);
}

// ---------------------------------------------------------------------------
// Fused GAT + GCN scatter: one wave per edge, 4 channels per lane.
//   g[d,c] += hlin[s,c] * coef_{head(c)}  +  hg[s,c] * (dinv[s]*w*dinv[d])
// ---------------------------------------------------------------------------
__global__ __launch_bounds__(256) void scatter_edges(
    const int* __restrict__ ei0, const int* __restrict__ ei1,
    const float* __restrict__ hlin, const float* __restrict__ hg,
    const float* __restrict__ sc_src, const float* __restrict__ sc_dst,
    const unsigned int* __restrict__ amax_u, const float* __restrict__ denom,
    const float* __restrict__ dinv, const float* __restrict__ ewp,
    float* __restrict__ g)
{
    int e = blockIdx.x * 8 + (threadIdx.x >> 5);
    if (e >= ETOT) return;
    int lane = threadIdx.x & 31;
    int s, d; float w;
    edge_decode(e, ei0, ei1, s, d, ewp[0], w);

    float nrm = dinv[s] * w * dinv[d];
    float coef[HEADS];
#pragma unroll
    for (int h = 0; h < HEADS; ++h) {
        float a = leaky(sc_src[s * HEADS + h] + sc_dst[d * HEADS + h]);
        float ex = expf(a - funflip(amax_u[d * HEADS + h]));
        coef[h] = ex / (denom[d * HEADS + h] + 1e-16f);
    }
    const float* __restrict__ hs = hlin + (size_t)s * HID;
    const float* __restrict__ gs = hg   + (size_t)s * HID;
    float* __restrict__ gd       = g    + (size_t)d * HID;
#pragma unroll
    for (int j = 0; j < HEADS; ++j) {
        int c = (j << 5) | lane;      // head(c) == j
        atomicAdd(&gd[c], hs[c] * coef[j] + gs[c] * nrm);
    }
}

// ---------------------------------------------------------------------------
// GraphNorm pieces
// ---------------------------------------------------------------------------
__global__ void gn_sum(const float* __restrict__ g, const int* __restrict__ batch,
                       float* __restrict__ gsum) {
    int t = blockIdx.x * blockDim.x + threadIdx.x;
    if (t >= N_NODES * HID) return;
    int n = t >> 7, c = t & (HID - 1);
    atomicAdd(&gsum[batch[n] * HID + c], g[t]);
}
// in place: p[t] /= max(cnt,1)   (used for mean and for variance)
__global__ void gn_div_cnt(float* __restrict__ p, const float* __restrict__ gcnt) {
    int t = blockIdx.x * blockDim.x + threadIdx.x;
    if (t >= G_GRAPHS * HID) return;
    p[t] /= fmaxf(gcnt[t >> 7], 1.0f);
}
// center in place + accumulate var sums
__global__ void gn_center(float* __restrict__ g, const int* __restrict__ batch,
                          const float* __restrict__ mean, const float* __restrict__ ms,
                          float* __restrict__ gvar) {
    int t = blockIdx.x * blockDim.x + threadIdx.x;
    if (t >= N_NODES * HID) return;
    int n = t >> 7, c = t & (HID - 1);
    float cen = g[t] - mean[batch[n] * HID + c] * ms[c];
    g[t] = cen;
    atomicAdd(&gvar[batch[n] * HID + c], cen * cen);
}
// h = elu(cen * rsqrt(var+eps) * w + b) [+ skipbuf]
__global__ void gn_apply(const float* __restrict__ g, const int* __restrict__ batch,
                         const float* __restrict__ gvar, const float* __restrict__ w,
                         const float* __restrict__ b, const float* __restrict__ skipbuf,
                         float* __restrict__ h) {
    int t = blockIdx.x * blockDim.x + threadIdx.x;
    if (t >= N_NODES * HID) return;
    int n = t >> 7, c = t & (HID - 1);
    float v = g[t] * rsqrtf(gvar[batch[n] * HID + c] + GN_EPS) * w[c] + b[c];
    float e = v > 0.0f ? v : (expf(v) - 1.0f);   // elu
    h[t] = skipbuf ? (e + skipbuf[t]) : e;
}

// ---------------------------------------------------------------------------
// Head: out[n] = sigmoid(p2[n,:64] @ w3 + b3)
// ---------------------------------------------------------------------------
__global__ void head_out(const float* __restrict__ p2, const float* __restrict__ w3,
                         const float* __restrict__ b3, float* __restrict__ out) {
    int n = blockIdx.x * blockDim.x + threadIdx.x;
    if (n >= N_NODES) return;
    float s = b3[0];
#pragma unroll
    for (int c = 0; c < HID / 2; ++c) s += p2[(size_t)n * (HID / 2) + c] * w3[c];
    out[n] = 1.0f / (1.0f + expf(-s));
}

// ---------------------------------------------------------------------------
// Host launcher
// ---------------------------------------------------------------------------
static inline int cdiv(long long a, long long b) { return (int)((a + b - 1) / b); }

extern "C" void kernel_launch(void* const* d_in, const int* in_sizes, int n_in,
                              void* d_out, int out_size, void* d_ws, size_t ws_size,
                              hipStream_t stream) {
    (void)in_sizes; (void)n_in; (void)out_size; (void)ws_size;

    const float* x       = (const float*)d_in[0];
    const int*   ei      = (const int*)d_in[1];
    const int*   ei0     = ei;
    const int*   ei1     = ei + E_EDGES;
    const int*   batch   = (const int*)d_in[2];
    const float* w_in1   = (const float*)d_in[3];
    const float* b_in1   = (const float*)d_in[4];
    const float* w_in2   = (const float*)d_in[5];
    const float* b_in2   = (const float*)d_in[6];
    const float* gat_w   = (const float*)d_in[7];
    const float* gat_as  = (const float*)d_in[8];
    const float* gat_ad  = (const float*)d_in[9];
    const float* gat_b   = (const float*)d_in[10];
    const float* gcn_w   = (const float*)d_in[11];
    const float* gcn_b   = (const float*)d_in[12];
    const float* norm_w  = (const float*)d_in[13];
    const float* norm_b  = (const float*)d_in[14];
    const float* norm_ms = (const float*)d_in[15];
    const float* skip_w  = (const float*)d_in[16];
    const float* skip_b  = (const float*)d_in[17];
    const float* w_h1    = (const float*)d_in[18];
    const float* b_h1    = (const float*)d_in[19];
    const float* w_h2    = (const float*)d_in[20];
    const float* b_h2    = (const float*)d_in[21];
    const float* w_h3    = (const float*)d_in[22];
    const float* b_h3    = (const float*)d_in[23];
    const float* ewp     = (const float*)d_in[24];

    // ---- workspace carve-up (floats) ----
    const size_t NH = (size_t)N_NODES * HID;
    float* ws    = (float*)d_ws;
    float* h     = ws;                 // N x HID
    float* hlin  = h    + NH;          // N x HID (GAT lin / skip / p1 scratch)
    float* hg    = hlin + NH;          // N x HID (GCN lin / p2 scratch)
    float* g     = hg   + NH;          // N x HID (conv accumulator / centered)
    float* sc_src = g + NH;                        // N x HEADS
    float* sc_dst = sc_src + (size_t)N_NODES * HEADS;
    unsigned int* amax_u = (unsigned int*)(sc_dst + (size_t)N_NODES * HEADS);
    float* denom = (float*)(amax_u + (size_t)N_NODES * HEADS);
    float* dinv  = denom + (size_t)N_NODES * HEADS;
    float* deg   = dinv + N_NODES;
    float* gsum  = deg + N_NODES;      // G x HID
    float* gvar  = gsum + G_GRAPHS * HID;
    float* gcnt  = gvar + G_GRAPHS * HID;  // G

    const int T = 256;
    const int gemmBlocks = N_NODES / 16;   // 3125, one 16-row tile per block

    // ---- input MLP ----
    gemm_wmma_f32_t<HID><<<gemmBlocks, T, 0, stream>>>(x,    w_in1, b_in1, hlin, 1);
    gemm_wmma_f32_t<HID><<<gemmBlocks, T, 0, stream>>>(hlin, w_in2, b_in2, h,    0);

    // ---- per-call graph invariants ----
    fill_f32<<<cdiv(N_NODES, T), T, 0, stream>>>(deg, 0.0f, N_NODES);
    fill_f32<<<1, 64, 0, stream>>>(gcnt, 0.0f, G_GRAPHS);
    edge_deg<<<cdiv(ETOT, T), T, 0, stream>>>(ei0, ei1, ewp, deg);
    node_dinv<<<cdiv(N_NODES, T), T, 0, stream>>>(deg, dinv);
    node_cnt<<<cdiv(N_NODES, T), T, 0, stream>>>(batch, gcnt);

    // ---- layers ----
    for (int i = 0; i < NLAYERS; ++i) {
        const float* gw  = gat_w  + (size_t)i * HID * HID;
        const float* cw  = gcn_w  + (size_t)i * HID * HID;
        const float* sw  = skip_w + (size_t)i * HID * HID;
        const float* as  = gat_as + (size_t)i * HEADS * HC;
        const float* ad  = gat_ad + (size_t)i * HEADS * HC;

        gemm_wmma_f32_t<HID><<<gemmBlocks, T, 0, stream>>>(h, gw, nullptr, hlin, 0);
        gat_scores<<<cdiv((long long)N_NODES * HEADS, T), T, 0, stream>>>(hlin, as, ad, sc_src, sc_dst);
        gemm_wmma_f32_t<HID><<<gemmBlocks, T, 0, stream>>>(h, cw, nullptr, hg, 0);

        init_g<<<cdiv(NH, T), T, 0, stream>>>(gat_b + (size_t)i * HID, gcn_b + (size_t)i * HID, g);
        fill_u32<<<cdiv((long long)N_NODES * HEADS, T), T, 0, stream>>>(amax_u, 0u, N_NODES * HEADS);
        fill_f32<<<cdiv((long long)N_NODES * HEADS, T), T, 0, stream>>>(denom, 0.0f, N_NODES * HEADS);

        edge_amax <<<cdiv((long long)ETOT * HEADS, T), T, 0, stream>>>(ei0, ei1, sc_src, sc_dst, amax_u);
        edge_denom<<<cdiv((long long)ETOT * HEADS, T), T, 0, stream>>>(ei0, ei1, sc_src, sc_dst, amax_u, denom);
        scatter_edges<<<cdiv(ETOT, 8), T, 0, stream>>>(ei0, ei1, hlin, hg, sc_src, sc_dst,
                                                       amax_u, denom, dinv, ewp, g);

        // skip branch (layers 0, 2): skipbuf (reuses hlin, free after scatter) = h @ skip_w + skip_b
        const bool do_skip = (i % 2 == 0);
        if (do_skip)
            gemm_wmma_f32_t<HID><<<gemmBlocks, T, 0, stream>>>(h, sw, skip_b + (size_t)i * HID,
                                                               hlin, 0);

        // GraphNorm
        fill_f32<<<cdiv(G_GRAPHS * HID, T), T, 0, stream>>>(gsum, 0.0f, G_GRAPHS * HID);
        fill_f32<<<cdiv(G_GRAPHS * HID, T), T, 0, stream>>>(gvar, 0.0f, G_GRAPHS * HID);
        gn_sum<<<cdiv(NH, T), T, 0, stream>>>(g, batch, gsum);
        gn_div_cnt<<<cdiv(G_GRAPHS * HID, T), T, 0, stream>>>(gsum, gcnt);               // -> mean
        gn_center<<<cdiv(NH, T), T, 0, stream>>>(g, batch, gsum, norm_ms + (size_t)i * HID, gvar);
        gn_div_cnt<<<cdiv(G_GRAPHS * HID, T), T, 0, stream>>>(gvar, gcnt);               // -> var
        gn_apply<<<cdiv(NH, T), T, 0, stream>>>(g, batch, gvar,
                                                norm_w + (size_t)i * HID, norm_b + (size_t)i * HID,
                                                do_skip ? hlin : nullptr, h);
    }

    // ---- head MLP ----
    gemm_wmma_f32_t<HID    ><<<gemmBlocks, T, 0, stream>>>(h,    w_h1, b_h1, hlin, 1);
    gemm_wmma_f32_t<HID / 2><<<gemmBlocks, T, 0, stream>>>(hlin, w_h2, b_h2, hg,   1);
    head_out<<<cdiv(N_NODES, T), T, 0, stream>>>(hg, w_h3, b_h3, (float*)d_out);
}